// Volume_45870250721617
// MI455X (gfx1250) — compile-verified
//
#include <hip/hip_runtime.h>
#include <math.h>

typedef float v2f __attribute__((ext_vector_type(2)));
typedef float v8f __attribute__((ext_vector_type(8)));

#define NVIEW 8
#define TILE  16
#define BASEDIM 128

// ds_swizzle_b32 group-of-32 XOR: newlane = lane ^ M   (offset = 0x1f | M<<10)
template <int IMM>
__device__ inline float swz(float v) {
    return __int_as_float(__builtin_amdgcn_ds_swizzle(__float_as_int(v), IMM));
}
#define SWZ_XOR1  0x041f
#define SWZ_XOR2  0x081f
#define SWZ_XOR4  0x101f
#define SWZ_XOR16 0x401f

// ---------- per-view projection matrix: P = intrs @ inv(c2w), rows 0..2 ----------
__global__ void proj_precompute(const float* __restrict__ intrs,
                                const float* __restrict__ c2ws,
                                float* __restrict__ Mout) {
    int v = threadIdx.x;
    if (v >= NVIEW) return;
    const float* K = intrs + v * 16;
    const float* M = c2ws + v * 16;
    // rigid inverse: w2c = [R^T | -R^T t]
    float w2c[12];
    for (int i = 0; i < 3; ++i)
        for (int j = 0; j < 3; ++j)
            w2c[i * 4 + j] = M[j * 4 + i];
    for (int i = 0; i < 3; ++i)
        w2c[i * 4 + 3] = -(M[0 * 4 + 3] * M[0 * 4 + i] +
                           M[1 * 4 + 3] * M[1 * 4 + i] +
                           M[2 * 4 + 3] * M[2 * 4 + i]);
    for (int i = 0; i < 3; ++i)
        for (int j = 0; j < 4; ++j) {
            float acc = K[i * 4 + 3] * ((j == 3) ? 1.0f : 0.0f);
            for (int k = 0; k < 3; ++k) acc += K[i * 4 + k] * w2c[k * 4 + j];
            Mout[v * 12 + i * 4 + j] = acc;
        }
}

// ---------- main fused kernel: 8 waves = 8 views, 16 points per block ----------
__global__ __launch_bounds__(256) void fuse_volume_kernel(
    const float* __restrict__ feat0,   // (8,4,512,512)
    const float* __restrict__ feat1,   // (8,4,256,256)
    const float* __restrict__ coords,  // (P,3)
    const float* __restrict__ Mproj,   // (8,3,4)
    const float* __restrict__ w1,      // (4,8)
    const float* __restrict__ b1,      // (8)
    const float* __restrict__ w2,      // (8)
    const float* __restrict__ b2,      // (1)
    const int*   __restrict__ stage_ptr,
    float* __restrict__ out,           // (P,8) feat_vol then (P,) mask
    int P)
{
    __shared__ float s_score[NVIEW][TILE];
    __shared__ float s_wf[NVIEW][TILE][4];
    __shared__ int   s_mask[NVIEW][TILE];

    const int tid   = threadIdx.x;
    const int wave  = __builtin_amdgcn_readfirstlane(tid >> 5);  // view index (wave-uniform)
    const int lane  = tid & 31;
    const int lp    = lane & 15;    // point within tile
    const int lvl   = lane >> 4;    // 0 -> feat0 (512), 1 -> feat1 (256)
    const int base  = blockIdx.x * TILE;
    const int p     = base + lp;
    const int stage = stage_ptr[0];

    float acc[4] = {0.f, 0.f, 0.f, 0.f};
    int maskv = 0;

    if (p < P) {
        const float voxel = 2.0f / (BASEDIM - 1);
        float wx = coords[p * 3 + 0] * voxel - 1.0f;
        float wy = coords[p * 3 + 1] * voxel - 1.0f;
        float wz = coords[p * 3 + 2] * voxel - 1.0f;
        const float* Pm = Mproj + wave * 12;   // wave-uniform -> SGPR loads
        float ix = Pm[0] * wx + Pm[1] * wy + Pm[2]  * wz + Pm[3];
        float iy = Pm[4] * wx + Pm[5] * wy + Pm[6]  * wz + Pm[7];
        float iz = Pm[8] * wx + Pm[9] * wy + Pm[10] * wz + Pm[11];
        float invz = 1.0f / iz;
        float nx = (ix * invz) * (1.0f / 127.5f) - 1.0f;  // (w-1)/2 with w=256
        float ny = (iy * invz) * (1.0f / 127.5f) - 1.0f;
        maskv = (fabsf(nx) <= 1.0f) && (fabsf(ny) <= 1.0f) && (iz > 0.0f);

        // ---- branch-free bilinear gather (zeros padding folded into weights) ----
        int   W    = lvl ? 256 : 512;
        float half = lvl ? 127.5f : 255.5f;     // (W-1)/2
        float px = (nx + 1.0f) * half;
        float py = (ny + 1.0f) * half;
        const float* img = (lvl ? feat1 : feat0) + (size_t)wave * 4 * W * W;

        float x0f = floorf(px), y0f = floorf(py);
        int x0 = (int)x0f, y0 = (int)y0f, x1 = x0 + 1, y1 = y0 + 1;
        float fx = px - x0f, fy = py - y0f;
        float w00 = (1.f - fy) * (1.f - fx), w01 = (1.f - fy) * fx;
        float w10 = fy * (1.f - fx),         w11 = fy * fx;

        bool use = !(lvl == 0 && stage == 1);       // stage 1 drops feat0
        // NaN px/py compare false -> inb false -> all weights selected to 0
        bool inb = (px > -1.0f) && (px < (float)W) &&
                   (py > -1.0f) && (py < (float)W) && use;
        bool vx0 = (x0 >= 0) && (x0 < W), vx1 = (x1 >= 0) && (x1 < W);
        bool vy0 = (y0 >= 0) && (y0 < W), vy1 = (y1 >= 0) && (y1 < W);
        float m00 = (inb && vx0 && vy0) ? w00 : 0.f;
        float m01 = (inb && vx1 && vy0) ? w01 : 0.f;
        float m10 = (inb && vx0 && vy1) ? w10 : 0.f;
        float m11 = (inb && vx1 && vy1) ? w11 : 0.f;

        int cx0 = min(max(x0, 0), W - 1), cx1 = min(max(x1, 0), W - 1);
        int cy0 = min(max(y0, 0), W - 1), cy1 = min(max(y1, 0), W - 1);
        int i00 = cy0 * W + cx0, i01 = cy0 * W + cx1;
        int i10 = cy1 * W + cx0, i11 = cy1 * W + cx1;
        int cs = W * W;
        #pragma unroll
        for (int c = 0; c < 4; ++c) {
            const float* bp = img + c * cs;
            acc[c] = bp[i00] * m00 + bp[i01] * m01 + bp[i10] * m10 + bp[i11] * m11;
        }
    }

    // sum the two pyramid levels across the lane halves (ds_swizzle xor-16)
    float s0 = acc[0] + swz<SWZ_XOR16>(acc[0]);
    float s1 = acc[1] + swz<SWZ_XOR16>(acc[1]);
    float s2 = acc[2] + swz<SWZ_XOR16>(acc[2]);
    float s3 = acc[3] + swz<SWZ_XOR16>(acc[3]);

    if (lane < 16 && p < P) {
        s_mask[wave][lp]  = maskv;
        s_wf[wave][lp][0] = s0;
        s_wf[wave][lp][1] = s1;
        s_wf[wave][lp][2] = s2;
        s_wf[wave][lp][3] = s3;
    }

    // ---- layer 1 via WMMA f32 16x16x4: A = warp_feat(16x4), B = w1 padded 4x16 ----
    // A layout: lanes 0-15 hold K=0,1 ; lanes 16-31 hold K=2,3 (VGPR0/VGPR1)
    v2f a, b;
    a.x = (lane < 16) ? s0 : s2;
    a.y = (lane < 16) ? s1 : s3;
    int n  = lp;                       // output column (hidden unit), cols 8-15 padded 0
    int nc = n & 7;                    // clamped index -> unconditional loads + cndmask
    int kb = (lane < 16) ? 0 : 2;      // B: same K half-split across lanes
    float w1a = w1[kb * 8 + nc];
    float w1b = w1[kb * 8 + 8 + nc];
    b.x = (n < 8) ? w1a : 0.f;
    b.y = (n < 8) ? w1b : 0.f;
    v8f cacc = {0.f, 0.f, 0.f, 0.f, 0.f, 0.f, 0.f, 0.f};
    cacc = __builtin_amdgcn_wmma_f32_16x16x4_f32(
        /*neg_a=*/false, a, /*neg_b=*/false, b,
        /*c_mod=*/(short)0, cacc, /*reuse_a=*/false, /*reuse_b=*/false);

    // ---- layer 2 (elu + dot with w2) directly on the D layout ----
    // D: VGPR j -> row j (lanes 0-15, col = lane) and row j+8 (lanes 16-31, col = lane-16)
    float b1v = b1[nc], w2v = w2[nc];
    float b1n = (n < 8) ? b1v : 0.f;
    float w2n = (n < 8) ? w2v : 0.f;
    float b2s = b2[0];
    #pragma unroll
    for (int j = 0; j < 8; ++j) {
        float h = cacc[j] + b1n;
        h = (h > 0.f) ? h : (expf(h) - 1.0f);   // elu
        float t = h * w2n;
        t += swz<SWZ_XOR1>(t);
        t += swz<SWZ_XOR2>(t);
        t += swz<SWZ_XOR4>(t);                  // sum over 8-lane group
        t += b2s;
        if ((lane & 15) == 0)                   // lane 0 -> row j, lane 16 -> row j+8
            s_score[wave][j + (lvl << 3)] = t;
    }
    __syncthreads();

    // ---- cross-view softmax + weighted mean/var (16 threads, one per point) ----
    if (tid < TILE) {
        int pp = base + tid;
        if (pp < P) {
            float xs[NVIEW];
            int   cnt = 0;
            float mx = -3.0e38f;
            #pragma unroll
            for (int v = 0; v < NVIEW; ++v) {
                int m = s_mask[v][tid];
                cnt += m;
                float s = m ? s_score[v][tid] : -1000000000.0f;
                xs[v] = s;
                mx = fmaxf(mx, s);
            }
            float se = 0.f;
            #pragma unroll
            for (int v = 0; v < NVIEW; ++v) { xs[v] = expf(xs[v] - mx); se += xs[v]; }
            float inv = 1.0f / se;
            #pragma unroll
            for (int c = 0; c < 4; ++c) {
                float mean = 0.f, sq = 0.f;
                #pragma unroll
                for (int v = 0; v < NVIEW; ++v) {
                    float fw = s_wf[v][tid][c] * xs[v] * inv;
                    mean += fw;
                    sq   += fw * fw;
                }
                out[(size_t)pp * 8 + c]     = mean;
                out[(size_t)pp * 8 + 4 + c] = sq - mean * mean;
            }
            out[(size_t)P * 8 + pp] = (cnt > 1) ? 1.0f : 0.0f;
        }
    }
}

extern "C" void kernel_launch(void* const* d_in, const int* in_sizes, int n_in,
                              void* d_out, int out_size, void* d_ws, size_t ws_size,
                              hipStream_t stream) {
    const float* feat0  = (const float*)d_in[0];
    const float* feat1  = (const float*)d_in[1];
    const float* coords = (const float*)d_in[2];
    const float* intrs  = (const float*)d_in[3];
    const float* c2ws   = (const float*)d_in[4];
    const float* w1     = (const float*)d_in[5];
    const float* b1     = (const float*)d_in[6];
    const float* w2     = (const float*)d_in[7];
    const float* b2     = (const float*)d_in[8];
    const int*   stage  = (const int*)d_in[9];
    float* out = (float*)d_out;

    int P = in_sizes[2] / 3;
    float* Mproj = (float*)d_ws;   // 8 * 12 floats

    proj_precompute<<<1, 32, 0, stream>>>(intrs, c2ws, Mproj);
    int nblk = (P + TILE - 1) / TILE;
    fuse_volume_kernel<<<nblk, 256, 0, stream>>>(feat0, feat1, coords, Mproj,
                                                 w1, b1, w2, b2, stage, out, P);
}